// Ctc_Layer_29360396435968
// MI455X (gfx1250) — compile-verified
//
#include <hip/hip_runtime.h>

// CTC forward loss on gfx1250 (MI455X).
// One workgroup (512 threads = 16 wave32) per batch sample; alpha recursion
// over L=513 extended states lives in LDS; the per-timestep gather of the
// 257 needed logits (blank + 256 label columns) is done with the CDNA5
// async global->LDS gather path (GLOBAL_LOAD_ASYNC_TO_LDS_B32, ASYNCcnt),
// pipelined 8 timesteps deep to hide HBM latency on the sequential scan.

#define K_N       128
#define K_T       1024
#define K_V       1024
#define K_S       256
#define K_L       (2 * K_S + 1)   /* 513 extended states */
#define K_THREADS 512
#define K_DEPTH   8               /* async prefetch depth (timesteps in flight) */
#define K_NEG     (-1e30f)

__device__ __forceinline__ unsigned lds_byte_off(const void* p) {
  // Generic pointers into LDS carry the wave-relative LDS byte offset in the
  // low 32 bits on gfx1250 (shared aperture lives in the high word), which is
  // exactly what GLOBAL_LOAD_ASYNC_TO_LDS wants in its VDST VGPR.
  return (unsigned)(unsigned long long)p;
}

__device__ __forceinline__ void async_gather_f32(const float* gsrc, unsigned lds_dst) {
  // Per-lane gather global -> LDS, tracked with ASYNCcnt (CDNA5).
  asm volatile("global_load_async_to_lds_b32 %0, %1, off"
               :
               : "v"(lds_dst), "v"(gsrc)
               : "memory");
}

__device__ __forceinline__ float logaddexp2f(float a, float b) {
  float m = fmaxf(a, b);
  return m + __logf(__expf(a - m) + __expf(b - m));
}

__global__ __launch_bounds__(K_THREADS) void ctc_alpha_kernel(
    const float* __restrict__ logits,          // [N, T, V]
    const int*   __restrict__ targets,         // [N, S]
    const int*   __restrict__ target_lengths,  // [N]
    float*       __restrict__ per_sample_loss) // [N]
{
  __shared__ int           s_lab[K_S];
  __shared__ unsigned char s_skip[K_L];
  __shared__ float         s_alpha[2][K_L + 2];     // ping-pong, 2-elem NEG pad
  __shared__ float         s_lp[K_DEPTH][K_S + 1];  // [slot][0]=blank, [slot][1+j]=label j

  const int n   = blockIdx.x;
  const int tid = threadIdx.x;
  const float* __restrict__ row_base = logits + (size_t)n * K_T * K_V;

  if (tid < K_S) s_lab[tid] = targets[(size_t)n * K_S + tid];
  if (tid < 2) { s_alpha[0][tid] = K_NEG; s_alpha[1][tid] = K_NEG; }
  __syncthreads();

  // skip transition allowed: odd s>=3 and tgt[j] != tgt[j-1]
  for (int s = tid; s < K_L; s += K_THREADS) {
    int allow = 0;
    if ((s & 1) && s >= 3) {
      int j = (s - 1) >> 1;
      allow = (s_lab[j] != s_lab[j - 1]) ? 1 : 0;
    }
    s_skip[s] = (unsigned char)allow;
  }

  // Gather assignment: thread 0 -> blank column, threads 1..256 -> label j-1.
  const bool gthread = (tid <= K_S);
  const int  gv = gthread ? ((tid == 0) ? 0 : s_lab[tid - 1]) : 0;

  // Prime the async pipeline: timesteps 0..DEPTH-1 in flight per issuing wave.
  if (gthread) {
    for (int t = 0; t < K_DEPTH; ++t) {
      async_gather_f32(row_base + (size_t)t * K_V + gv,
                       lds_byte_off(&s_lp[t][tid]));
    }
  }

  for (int t = 0; t < K_T; ++t) {
    // Oldest outstanding gather (for this timestep) has landed in LDS.
    asm volatile("s_wait_asynccnt 7" ::: "memory");  // K_DEPTH - 1
    __syncthreads();

    const float* lp = s_lp[t & (K_DEPTH - 1)];
    float* Aw = s_alpha[t & 1];
    if (t == 0) {
      for (int s = tid; s < K_L; s += K_THREADS) {
        float a = K_NEG;
        if (s == 0)      a = lp[0];
        else if (s == 1) a = lp[1];
        Aw[2 + s] = a;
      }
    } else {
      const float* Ar = s_alpha[(t - 1) & 1];
      for (int s = tid; s < K_L; s += K_THREADS) {
        float a0 = Ar[2 + s];        // stay
        float a1 = Ar[1 + s];        // s-1 (pad -> NEG)
        float r  = logaddexp2f(a0, a1);
        if (s_skip[s]) r = logaddexp2f(r, Ar[s]);   // s-2
        float lpv = (s & 1) ? lp[1 + ((s - 1) >> 1)] : lp[0];
        Aw[2 + s] = r + lpv;
      }
    }
    __syncthreads();  // all reads of ring slot (t % DEPTH) and Ar are done

    const int tn = t + K_DEPTH;
    if (tn < K_T && gthread) {
      async_gather_f32(row_base + (size_t)tn * K_V + gv,
                       lds_byte_off(&s_lp[tn & (K_DEPTH - 1)][tid]));
    }
  }

  if (tid == 0) {
    const float* A = s_alpha[(K_T - 1) & 1];
    int len = target_lengths[n];
    int e = 2 * len;                       // last blank; e-1 = last label
    float ll = logaddexp2f(A[2 + e], A[2 + e - 1]);
    per_sample_loss[n] = -ll / (float)len; // CTC 'mean': divide by target len
  }
}

__global__ void ctc_reduce_kernel(const float* __restrict__ per,
                                  float* __restrict__ out) {
  __shared__ float sm[K_N];
  int tid = threadIdx.x;
  sm[tid] = per[tid];
  __syncthreads();
  for (int off = K_N / 2; off > 0; off >>= 1) {
    if (tid < off) sm[tid] += sm[tid + off];
    __syncthreads();
  }
  if (tid == 0) out[0] = sm[0] * (1.0f / (float)K_N);
}

extern "C" void kernel_launch(void* const* d_in, const int* in_sizes, int n_in,
                              void* d_out, int out_size, void* d_ws, size_t ws_size,
                              hipStream_t stream) {
  (void)in_sizes; (void)n_in; (void)out_size; (void)ws_size;
  const float* logits         = (const float*)d_in[0];
  const int*   targets        = (const int*)d_in[1];
  // d_in[2] = input_lengths: all == T in the reference module, unused.
  const int*   target_lengths = (const int*)d_in[3];

  float* per_sample = (float*)d_ws;  // 128 floats of scratch

  ctc_alpha_kernel<<<K_N, K_THREADS, 0, stream>>>(logits, targets,
                                                  target_lengths, per_sample);
  ctc_reduce_kernel<<<1, K_N, 0, stream>>>(per_sample, (float*)d_out);
}